// SuperLoss_4097398800592
// MI455X (gfx1250) — compile-verified
//
#include <hip/hip_runtime.h>
#include <stdint.h>

// ---------------------------------------------------------------------------
// SuperLoss on MI455X (gfx1250).
// Memory-bound streaming problem (~1.05 GB read, ~45us floor @ 23.3 TB/s).
// Data path: Tensor Data Mover (TDM) DMA of 2-D tiles (8 rows x 1000 f32)
// global -> LDS; logits tile issued by wave 0, targets tile by wave 1 so the
// two per-WGP TDM engines run concurrently; tracked by TENSORcnt.
// Compute reads LDS b128; wave32 shfl_xor reductions; deterministic final
// reduction kernel (no float atomics).
// ---------------------------------------------------------------------------

#define C_CLASSES 1000
#define NV4       250          // float4s per row (1000/4); row = 4000B, 16B aligned
#define RPB       8            // rows per block (1 row per wave32, 256 threads)
#define TAU_F     6.90775527898213705f     // log(1000)
#define LAM_F     1.0f
#define NEG2INVE  (-0.735758882342884644f) // -2/e
#define E_F       2.71828182845904523f

typedef unsigned int u32;
typedef u32 u32x4 __attribute__((ext_vector_type(4)));
typedef int i32x8 __attribute__((ext_vector_type(8)));
typedef int i32x4 __attribute__((ext_vector_type(4)));

#if defined(__has_builtin)
#if __has_builtin(__builtin_amdgcn_tensor_load_to_lds) && \
    __has_builtin(__builtin_amdgcn_s_wait_tensorcnt)
#define USE_TDM 1
#endif
#endif
#ifndef USE_TDM
#define USE_TDM 0
#endif

// LDS byte offset of a generic pointer into __shared__ (AS3 ptrtoint = offset).
__device__ static inline u32 lds_off(const void* p) {
  return (u32)(uintptr_t)(__attribute__((address_space(3))) const char*)p;
}

#if USE_TDM
// Issue one TDM descriptor: load `rows` x C_CLASSES f32 tile (row-major,
// stride C_CLASSES) from `gsrc` into LDS at byte offset `ldsb`.
// D# layout per CDNA5 ISA 8.3 (group0) / 8.4 (group1); groups 2/3 zero (2-D).
__device__ static inline void tdm_load_tile(const float* gsrc, u32 ldsb, u32 rows) {
  const uint64_t ga = (uint64_t)(uintptr_t)gsrc;
  u32x4 g0;
  g0.x = 1u;                                   // count=1 (valid), user mode, no gather
  g0.y = ldsb;                                 // lds_addr [63:32]
  g0.z = (u32)(ga & 0xFFFFFFFFu);              // global_addr[31:0]
  g0.w = (u32)((ga >> 32) & 0x01FFFFFFu)       // global_addr[56:32]
       | (2u << 30);                           // type=2 ("image")
  const u32 td0 = C_CLASSES, td1 = rows;       // tensor dims (OOB bounds)
  const u32 tl0 = C_CLASSES, tl1 = rows;       // tile dims
  const u32 st0 = C_CLASSES;                   // tensor_dim0_stride (elements)
  i32x8 g1;
  g1[0] = (int)(2u << 16);                     // data_size=2 -> 4B; mask/flags/pad=0
  g1[1] = (int)((td0 & 0xFFFFu) << 16);        // [63:48] tensor_dim0 lo16
  g1[2] = (int)((td0 >> 16) | ((td1 & 0xFFFFu) << 16)); // dim0 hi16 | dim1 lo16
  g1[3] = (int)((td1 >> 16) | (tl0 << 16));    // dim1 hi16 | tile_dim0
  g1[4] = (int)(tl1);                          // tile_dim1 | tile_dim2(=0)<<16
  g1[5] = (int)st0;                            // tensor_dim0_stride[31:0]
  g1[6] = 0;                                   // stride0 hi16 | stride1 lo16
  g1[7] = 0;                                   // stride1 hi32
  i32x4 z4 = {0, 0, 0, 0};
  i32x8 z8 = {0, 0, 0, 0, 0, 0, 0, 0};
  __builtin_amdgcn_tensor_load_to_lds(g0, g1, z4, z4, z8, 0);
}
#endif

__global__ __launch_bounds__(256) void superloss_rows_kernel(
    const float* __restrict__ logits, const float* __restrict__ targets,
    float* __restrict__ row_out) {
  // [ 8 rows of logits | 8 rows of targets ] = 64000 B static LDS
  __shared__ __align__(16) float smem[RPB * 2 * C_CLASSES];
  float* sL = smem;
  float* sT = smem + RPB * C_CLASSES;

  const int wave = (int)(threadIdx.x >> 5);
  const int lane = (int)(threadIdx.x & 31);
  const int row0 = (int)blockIdx.x * RPB;

#if USE_TDM
  // Two DMAs from two different waves -> both per-WGP TDM engines in flight.
  if (wave == 0) {
    tdm_load_tile(logits + (size_t)row0 * C_CLASSES, lds_off(sL), RPB);
    __builtin_amdgcn_s_wait_tensorcnt(0);       // wave 0's TENSORcnt == 0
  } else if (wave == 1) {
    tdm_load_tile(targets + (size_t)row0 * C_CLASSES, lds_off(sT), RPB);
    __builtin_amdgcn_s_wait_tensorcnt(0);       // wave 1's TENSORcnt == 0
  }
  __syncthreads();
#else
  {   // fallback: cooperative b128 copy global -> LDS
    const float4* gl4 = (const float4*)(logits  + (size_t)row0 * C_CLASSES);
    const float4* gt4 = (const float4*)(targets + (size_t)row0 * C_CLASSES);
    float4* sl4 = (float4*)sL;
    float4* st4 = (float4*)sT;
    for (int i = (int)threadIdx.x; i < RPB * NV4; i += 256) {
      sl4[i] = gl4[i];
      st4[i] = gt4[i];
    }
    __syncthreads();
  }
#endif

  // ---- per-wave row processing from LDS (b128 reads) ----
  const float4* l4 = (const float4*)(sL + wave * C_CLASSES);
  const float4* t4 = (const float4*)(sT + wave * C_CLASSES);

  // pass 1: row max
  float m = -__FLT_MAX__;
#pragma unroll
  for (int k = 0; k < 8; ++k) {
    const int idx = lane + 32 * k;
    if (idx < NV4) {
      const float4 v = l4[idx];
      m = fmaxf(m, fmaxf(fmaxf(v.x, v.y), fmaxf(v.z, v.w)));
    }
  }
#pragma unroll
  for (int off = 16; off; off >>= 1) m = fmaxf(m, __shfl_xor(m, off, 32));

  // pass 2: sum exp(l-m), sum t*l, sum |l-t|
  float se = 0.f, stl = 0.f, sab = 0.f;
#pragma unroll
  for (int k = 0; k < 8; ++k) {
    const int idx = lane + 32 * k;
    if (idx < NV4) {
      const float4 l = l4[idx];
      const float4 t = t4[idx];
      se  += __expf(l.x - m) + __expf(l.y - m) + __expf(l.z - m) + __expf(l.w - m);
      stl += l.x * t.x + l.y * t.y + l.z * t.z + l.w * t.w;
      sab += fabsf(l.x - t.x) + fabsf(l.y - t.y) + fabsf(l.z - t.z) + fabsf(l.w - t.w);
    }
  }
#pragma unroll
  for (int off = 16; off; off >>= 1) {
    se  += __shfl_xor(se,  off, 32);
    stl += __shfl_xor(stl, off, 32);
    sab += __shfl_xor(sab, off, 32);
  }

  if (lane == 0) {
    // l_i = -sum t*log_softmax = m + logZ - sum(t*l)  (rows of t sum to 1)
    const float l_i = m + logf(se) - stl;
    const float y = 0.5f * fmaxf(NEG2INVE, (l_i - TAU_F) / LAM_F);
    // Lambert W0 via Halley (exact port of reference, incl. 1e-12 guards)
    const float p = sqrtf(fmaxf(2.f * (E_F * y + 1.f), 0.f));
    float w = (y < -0.25f) ? (-1.f + p) : log1pf(fmaxf(y, -0.25f));
#pragma unroll
    for (int it = 0; it < 12; ++it) {
      const float ew = expf(w);
      const float f = w * ew - y;
      float denom = ew * (w + 1.f) - (w + 2.f) * f / (2.f * w + 2.f + 1e-12f);
      denom = (fabsf(denom) < 1e-12f) ? 1e-12f : denom;
      const float step = f / denom;
      w = (fabsf(f) < 1e-12f) ? w : (w - step);
    }
    const float sigma = expf(-w);
    const float logsig = -w;  // log(sigma)
    // sum_j (|l-t| - tau)*sigma + lam*logsig^2  over C columns
    const float rowv = (sab - (float)C_CLASSES * TAU_F) * sigma +
                       LAM_F * logsig * logsig * (float)C_CLASSES;
    row_out[row0 + wave] = rowv;
  }
}

// Deterministic final reduction: fixed traversal order, single block.
__global__ __launch_bounds__(1024) void superloss_reduce_kernel(
    const float* __restrict__ row_vals, float* __restrict__ out, int n) {
  __shared__ float red[32];
  float s = 0.f;
  for (int i = (int)threadIdx.x; i < n; i += 1024) s += row_vals[i];
#pragma unroll
  for (int off = 16; off; off >>= 1) s += __shfl_xor(s, off, 32);
  const int wave = (int)(threadIdx.x >> 5);
  const int lane = (int)(threadIdx.x & 31);
  if (lane == 0) red[wave] = s;
  __syncthreads();
  if (wave == 0) {
    s = red[lane];
#pragma unroll
    for (int off = 16; off; off >>= 1) s += __shfl_xor(s, off, 32);
    if (lane == 0) out[0] = s * (1.0f / 131072.0f);  // / BATCH_SIZE
  }
}

extern "C" void kernel_launch(void* const* d_in, const int* in_sizes, int n_in,
                              void* d_out, int out_size, void* d_ws, size_t ws_size,
                              hipStream_t stream) {
  const float* logits  = (const float*)d_in[0];
  const float* targets = (const float*)d_in[1];
  float* rowbuf = (float*)d_ws;                 // N floats of per-row partials
  const int n_rows = in_sizes[0] / C_CLASSES;   // 131072

  superloss_rows_kernel<<<n_rows / RPB, 256, 0, stream>>>(logits, targets, rowbuf);
  superloss_reduce_kernel<<<1, 1024, 0, stream>>>(rowbuf, (float*)d_out, n_rows);
}